// Mode_att_7404523618910
// MI455X (gfx1250) — compile-verified
//
#include <hip/hip_runtime.h>
#include <hip/hip_bf16.h>
#include <stdint.h>

// Problem constants from the reference
#define Bq   64
#define Nq   883
#define Tq   12
#define TNq  16
#define TODq 288
#define WPB  8   // waves per block (wave32); each wave handles TWO (b,n) pairs

// Raw LDS byte offset (addrspace(3) value) from a generic shared pointer.
__device__ __forceinline__ uint32_t lds_off(const void* p) {
  return (uint32_t)(uintptr_t)((__attribute__((address_space(3))) const void*)p);
}

extern "C" __global__ __launch_bounds__(WPB * 32)
void modeatt_kernel(const float* __restrict__ enc,
                    const int*   __restrict__ xmark,
                    const float* __restrict__ dec,
                    const float* __restrict__ kbank,
                    const float* __restrict__ vbank,
                    const float* __restrict__ attw,
                    const float* __restrict__ attb,
                    float* __restrict__ out)
{
  // per wave, per half: [0..191]=K tile, [192..383]=V tile
  __shared__ float kv[WPB][2][2 * TNq * Tq];   // 24576 B
  __shared__ float ash[WPB][2][TNq];           // 1024 B

  const int wave = threadIdx.x >> 5;
  const int lane = threadIdx.x & 31;
  const int half = lane >> 4;        // which (b,n) pair this lane works on
  const int y    = lane & 15;        // K-row index within the pair
  const int b    = blockIdx.y;

  const int n0 = blockIdx.x * (2 * WPB) + wave * 2;
  if (n0 >= Nq) return;                               // wave-uniform early-out
  const int  n1c = (n0 + 1 < Nq) ? (n0 + 1) : (Nq - 1); // clamp for safe DMA
  const int  nh  = half ? n1c : n0;
  const bool act = half ? (n0 + 1 < Nq) : true;

  const int t = xmark[b * 2];  // x_mark_enc[b,0,0]

  const size_t off0 = ((size_t)n0  * TODq + (size_t)t) * (size_t)(TNq * Tq);
  const size_t off1 = ((size_t)n1c * TODq + (size_t)t) * (size_t)(TNq * Tq);

  // ---- async DMA: 4 tiles x 768B into LDS; 12x B64 ops (256B each, all lanes) ----
  {
    uint32_t lk0 = lds_off(&kv[wave][0][0]) + (uint32_t)(lane * 8);
    uint32_t lk1 = lds_off(&kv[wave][1][0]) + (uint32_t)(lane * 8);
    uint32_t lv0 = lk0 + 768u;
    uint32_t lv1 = lk1 + 768u;
    uint64_t gk0 = (uint64_t)(uintptr_t)(kbank + off0) + (uint64_t)(lane * 8);
    uint64_t gk1 = (uint64_t)(uintptr_t)(kbank + off1) + (uint64_t)(lane * 8);
    uint64_t gv0 = (uint64_t)(uintptr_t)(vbank + off0) + (uint64_t)(lane * 8);
    uint64_t gv1 = (uint64_t)(uintptr_t)(vbank + off1) + (uint64_t)(lane * 8);
    asm volatile(
        "global_load_async_to_lds_b64 %0, %4, off\n\t"
        "global_load_async_to_lds_b64 %0, %4, off offset:256\n\t"
        "global_load_async_to_lds_b64 %0, %4, off offset:512\n\t"
        "global_load_async_to_lds_b64 %1, %5, off\n\t"
        "global_load_async_to_lds_b64 %1, %5, off offset:256\n\t"
        "global_load_async_to_lds_b64 %1, %5, off offset:512\n\t"
        "global_load_async_to_lds_b64 %2, %6, off\n\t"
        "global_load_async_to_lds_b64 %2, %6, off offset:256\n\t"
        "global_load_async_to_lds_b64 %2, %6, off offset:512\n\t"
        "global_load_async_to_lds_b64 %3, %7, off\n\t"
        "global_load_async_to_lds_b64 %3, %7, off offset:256\n\t"
        "global_load_async_to_lds_b64 %3, %7, off offset:512"
        :
        : "v"(lk0), "v"(lk1), "v"(lv0), "v"(lv1),
          "v"(gk0), "v"(gk1), "v"(gv0), "v"(gv1)
        : "memory");
  }

  // ---- overlap with DMA: load Q (uniform per half) and gate params ----
  const size_t bnh = (size_t)b * Nq + (size_t)nh;
  const float4* qv4 = (const float4*)(enc + bnh * Tq);
  float4 qa = qv4[0], qb = qv4[1], qc = qv4[2];
  float q[12] = {qa.x, qa.y, qa.z, qa.w, qb.x, qb.y, qb.z, qb.w,
                 qc.x, qc.y, qc.z, qc.w};

  const float aw    = attw[nh * 17 + y];
  const float awQ   = attw[nh * 17 + 16];
  const float abias = attb[nh];

  asm volatile("s_wait_asynccnt 0" ::: "memory");

  // ---- Euclidean distance for K row y of this half's tile ----
  const float* Kr = &kv[wave][half][y * Tq];
  float d2 = 0.f;
#pragma unroll
  for (int d = 0; d < Tq; ++d) {
    float df = q[d] - Kr[d];
    d2 = fmaf(df, df, d2);
  }
  const float dist = __builtin_amdgcn_sqrtf(d2);

  // ---- reduction pass A: sum, sum of squares, min (per 16-lane half) ----
  float s1 = dist, s2 = dist * dist, dmin = dist;
#pragma unroll
  for (int m = 1; m <= 8; m <<= 1) {
    s1   += __shfl_xor(s1, m, 32);
    s2   += __shfl_xor(s2, m, 32);
    dmin  = fminf(dmin, __shfl_xor(dmin, m, 32));
  }

  // 16-way z-score + softmax (ddof=1); max score = (mean - min_dist)*scale
  const float m16   = s1 * (1.f / 16.f);
  const float v16   = fmaxf((s2 - 16.f * m16 * m16) * (1.f / 15.f), 0.f);
  const float inv16 = 10.f * __builtin_amdgcn_rcpf(__builtin_amdgcn_sqrtf(v16) + 1e-6f);
  const float mx16  = (m16 - dmin) * inv16;
  const float e16   = __expf((m16 - dist) * inv16 - mx16);

  // 17-way: extra row is Q+0.1 -> its distance is the constant 0.1*sqrt(12)
  const float cQ    = 0.34641016151377545870f;
  const float m17   = (s1 + cQ) * (1.f / 17.f);
  const float v17   = fmaxf((s2 + cQ * cQ - 17.f * m17 * m17) * (1.f / 16.f), 0.f);
  const float inv17 = 10.f * __builtin_amdgcn_rcpf(__builtin_amdgcn_sqrtf(v17) + 1e-6f);
  const float scQ   = (m17 - cQ) * inv17;
  const float mx17  = fmaxf((m17 - dmin) * inv17, scQ);
  const float e17   = __expf((m17 - dist) * inv17 - mx17);
  const float eQ    = __expf(scQ - mx17);

  // ---- reduction pass B: softmax denominators + gate dot-product numerator ----
  float r16 = e16, r17 = e17, rga = e17 * aw;
#pragma unroll
  for (int m = 1; m <= 8; m <<= 1) {
    r16 += __shfl_xor(r16, m, 32);
    r17 += __shfl_xor(r17, m, 32);
    rga += __shfl_xor(rga, m, 32);
  }

  const float att = e16 * __builtin_amdgcn_rcpf(r16);
  const float w   = (rga + eQ * awQ) * __builtin_amdgcn_rcpf(r17 + eQ) + abias;

  // publish att: every lane writes its unique (half, y) slot
  ash[wave][half][y] = att;

  // lanes y<12 of each half: att_out[d] = sum_y att[y]*V[y][d]; blend; store
  if (y < 12 && act) {
    const float* Vt = &kv[wave][half][TNq * Tq];
    float acc = 0.f;
#pragma unroll
    for (int yy = 0; yy < TNq; ++yy)
      acc = fmaf(ash[wave][half][yy], Vt[yy * Tq + y], acc);
    const size_t oidx = bnh * Tq + (size_t)y;
    out[oidx] = (1.f - w) * acc + w * dec[oidx];
  }
}

extern "C" void kernel_launch(void* const* d_in, const int* in_sizes, int n_in,
                              void* d_out, int out_size, void* d_ws, size_t ws_size,
                              hipStream_t stream) {
  const float* enc = (const float*)d_in[0];
  const int*   xm  = (const int*)d_in[1];
  const float* dc  = (const float*)d_in[2];
  const float* kb  = (const float*)d_in[3];
  const float* vb  = (const float*)d_in[4];
  const float* aw  = (const float*)d_in[5];
  const float* ab  = (const float*)d_in[6];
  float* out = (float*)d_out;

  dim3 grid((Nq + 2 * WPB - 1) / (2 * WPB), Bq);
  modeatt_kernel<<<grid, WPB * 32, 0, stream>>>(enc, xm, dc, kb, vb, aw, ab, out);

  (void)in_sizes; (void)n_in; (void)out_size; (void)d_ws; (void)ws_size;
}